// Gradient_img_9088150798663
// MI455X (gfx1250) — compile-verified
//
#include <hip/hip_runtime.h>
#include <math.h>

// Sobel gradient-magnitude stencil, MI455X (gfx1250), memory-bound.
// Data path: async global->LDS tile staging (ASYNCcnt), ds_load compute reads,
// separable 3x3 arithmetic, aligned float2 stores.

#define IMG_H 512
#define IMG_W 512
#define OUT_H 510
#define OUT_W 510

#define TILE_OW 128            // output tile width
#define TILE_OH 32             // output tile height
#define LDS_STRIDE 132         // 33 x float4 chunks per row (covers 130 needed cols)
#define IN_H 34                // TILE_OH + 2 halo rows
#define CHUNKS_PER_ROW 33
#define TOTAL_CHUNKS (IN_H * CHUNKS_PER_ROW)   // 1122 B128 chunks per tile

__global__ __launch_bounds__(256)
void sobel_mag_kernel(const float* __restrict__ im, float* __restrict__ out)
{
    __shared__ float tile[IN_H * LDS_STRIDE];  // 17,952 B of 320 KB/WGP

    // --- Escape hatch: the async-copy asm below addresses `tile` only via raw
    // integer LDS offsets, which LLVM cannot see. Without this capture, `tile`
    // is a never-stored, non-escaping global and every ds_load of it folds to
    // poison (observed in rounds 1-2: compute collapsed, ds=0). Passing the
    // pointer through an empty asm marks it escaped, so the later "memory"
    // clobbers are assumed to write it. Emits zero instructions.
    {
        float* tileEsc = &tile[0];
        asm volatile("" : "+v"(tileEsc) :: "memory");
    }

    const int tid  = threadIdx.x;
    const int tX   = blockIdx.x;   // 0..3
    const int tY   = blockIdx.y;   // 0..15
    const int img  = blockIdx.z;   // 0..B*C-1

    const int col0 = tX * TILE_OW; // input & output column origin
    const int row0 = tY * TILE_OH;

    const unsigned imgBase = (unsigned)img * (unsigned)(IMG_H * IMG_W);

    // ---- Stage the (132 x 34) fp32 input tile into LDS via async B128 copies ----
    for (int t = tid; t < TOTAL_CHUNKS; t += 256) {
        int r  = t / CHUNKS_PER_ROW;
        int cc = (t - r * CHUNKS_PER_ROW) * 4;      // local col of this 4-float chunk
        int gr = row0 + r;  if (gr > IMG_H - 1) gr = IMG_H - 1;   // clamped rows never read
        int gc = col0 + cc; if (gc + 4 > IMG_W) gc = col0;        // redirected chunks never read
        unsigned gOff = (imgBase + (unsigned)gr * (unsigned)IMG_W + (unsigned)gc) * 4u;
        unsigned lOff = ((unsigned)r * (unsigned)LDS_STRIDE + (unsigned)cc) * 4u;
        // GVS mode: mem = SADDR(64b) + VADDR(32b per-lane) ; VDST = LDS byte address
        asm volatile("global_load_async_to_lds_b128 %0, %1, %2 offset:0"
                     :
                     : "v"(lOff), "v"(gOff), "s"(im)
                     : "memory");
    }
    asm volatile("s_wait_asynccnt 0" ::: "memory");
    __syncthreads();

    // ---- Compute: each thread -> 4 cols x 4 rows = 16 outputs ----
    const int cg = tid & 31;        // column group (0..31)
    const int rg = tid >> 5;        // row group    (0..7)
    const int cl = cg * 4;          // local output col (0..124)

    #pragma unroll
    for (int k = 0; k < 4; ++k) {
        const int rl    = rg + k * 8;                 // local output row (0..31)
        const int base0 = (rl + 0) * LDS_STRIDE + cl; // 16B-aligned (cl % 4 == 0)
        const int base1 = (rl + 1) * LDS_STRIDE + cl;
        const int base2 = (rl + 2) * LDS_STRIDE + cl;

        float s[6], d[6];
        #pragma unroll
        for (int j = 0; j < 6; ++j) {
            // Direct indexing on the __shared__ variable -> addrspace(3) ds_loads.
            const float a = tile[base0 + j];
            const float b = tile[base1 + j];
            const float c = tile[base2 + j];
            s[j] = a + 2.0f * b + c;   // column smoothing (for gx)
            d[j] = a - c;              // column difference (for gy)
        }

        float m[4];
        #pragma unroll
        for (int o = 0; o < 4; ++o) {
            const float gx = s[o] - s[o + 2];
            const float gy = d[o] + 2.0f * d[o + 1] + d[o + 2];
            m[o] = 0.5f * sqrtf(gx * gx + gy * gy);
        }

        const int oy = row0 + rl;
        const int ox = col0 + cl;
        if (oy < OUT_H) {
            float* op = out + (size_t)img * (OUT_H * OUT_W) + (size_t)oy * OUT_W + ox;
            if (ox + 4 <= OUT_W) {
                // row base index is even -> 8-byte aligned; 510-wide rows forbid 16B alignment
                ((float2*)op)[0] = make_float2(m[0], m[1]);
                ((float2*)op)[1] = make_float2(m[2], m[3]);
            } else {
                #pragma unroll
                for (int o = 0; o < 4; ++o)
                    if (ox + o < OUT_W) op[o] = m[o];
            }
        }
    }
}

extern "C" void kernel_launch(void* const* d_in, const int* in_sizes, int n_in,
                              void* d_out, int out_size, void* d_ws, size_t ws_size,
                              hipStream_t stream)
{
    (void)n_in; (void)out_size; (void)d_ws; (void)ws_size;
    const float* im  = (const float*)d_in[0];
    float*       out = (float*)d_out;
    const int images = in_sizes[0] / (IMG_H * IMG_W);   // B*C = 176

    dim3 grid((OUT_W + TILE_OW - 1) / TILE_OW,          // 4
              (OUT_H + TILE_OH - 1) / TILE_OH,          // 16
              images);                                  // 176
    sobel_mag_kernel<<<grid, 256, 0, stream>>>(im, out);
}